// GraphEncoder_1288490189296
// MI455X (gfx1250) — compile-verified
//
#include <hip/hip_runtime.h>

#define N_NODES 100000
#define N_EDGES 1600000
#define HIDDEN  128

typedef __attribute__((ext_vector_type(2))) float v2f;
typedef __attribute__((ext_vector_type(8))) float v8f;

// ---------------- small utility kernels ----------------

__global__ void zero_f32(float* __restrict__ p, int n) {
    int i = blockIdx.x * blockDim.x + threadIdx.x;
    if (i < n) p[i] = 0.0f;
}

// deg[dst] += 1 for every edge (in-degree at target nodes)
__global__ void degree_kernel(const int* __restrict__ dst, float* __restrict__ deg, int E) {
    int e = blockIdx.x * blockDim.x + threadIdx.x;
    if (e < E) atomicAdd(&deg[dst[e]], 1.0f);
}

// in-place: deg -> deg>0 ? rsqrt(deg) : 0
__global__ void dinv_kernel(float* __restrict__ deg, int n) {
    int i = blockIdx.x * blockDim.x + threadIdx.x;
    if (i < n) {
        float d = deg[i];
        deg[i] = (d > 0.0f) ? rsqrtf(d) : 0.0f;
    }
}

// out[n, d] = bias[d]  (accumulator initialized with the layer bias)
__global__ void bias_init_kernel(const float* __restrict__ bias, float* __restrict__ out, int total) {
    int i = blockIdx.x * blockDim.x + threadIdx.x;
    if (i < total) out[i] = bias[i & (HIDDEN - 1)];
}

// ---------------- fp32 WMMA GEMM: H = X (N x 128) * W (128 x 128) ----------------
// grid.x = N/16 row tiles; 256 threads = 8 waves; wave w owns column tile w (16 cols).
// Each wave: 16x16 f32 output tile, K stepped by 4 via V_WMMA_F32_16X16X4_F32.
__global__ void __launch_bounds__(256)
gemm_wmma_f32(const float* __restrict__ X, const float* __restrict__ W, float* __restrict__ H) {
    const int lane   = threadIdx.x & 31;
    const int wave   = threadIdx.x >> 5;   // 0..7 -> column tile
    const int laneLo = lane & 15;          // 0..15
    const int laneHi = lane >> 4;          // 0 or 1

    const int row0 = blockIdx.x * 16;
    const int col0 = wave * 16;

    // A fragment source: row M = row0 + laneLo; K offsets 2*laneHi + {0,1}
    const float* __restrict__ xr = X + (size_t)(row0 + laneLo) * HIDDEN + 2 * laneHi;
    // B fragment source: col N = col0 + laneLo; K rows 2*laneHi + {0,1}
    const float* __restrict__ wc = W + col0 + laneLo;

    v8f acc = {};
#pragma unroll 8
    for (int k0 = 0; k0 < HIDDEN; k0 += 4) {
        v2f a, b;
        a.x = xr[k0 + 0];
        a.y = xr[k0 + 1];
        const int kb = k0 + 2 * laneHi;
        b.x = wc[(size_t)(kb + 0) * HIDDEN];
        b.y = wc[(size_t)(kb + 1) * HIDDEN];
        // 8 args: (neg_a, A, neg_b, B, c_mod, C, reuse_a, reuse_b)
        acc = __builtin_amdgcn_wmma_f32_16x16x4_f32(
            false, a, false, b, (short)0, acc, false, false);
    }

    // D layout: VGPR v -> row M = v + 8*laneHi, col N = laneLo
    float* __restrict__ hr = H + (size_t)row0 * HIDDEN + col0 + laneLo;
#pragma unroll
    for (int v = 0; v < 8; ++v) {
        hr[(size_t)(v + 8 * laneHi) * HIDDEN] = acc[v];
    }
}

// ---------------- edge scatter: out[dst,:] += h[src,:] * dinv[src]*dinv[dst] ----------------
// one wave32 per edge; lane carries 4 consecutive features (float4 gather, 4 f32 atomics)
__global__ void __launch_bounds__(256)
scatter_kernel(const float* __restrict__ H, const int* __restrict__ src,
               const int* __restrict__ dst, const float* __restrict__ dinv,
               float* __restrict__ out, int E) {
    const int gt   = blockIdx.x * blockDim.x + threadIdx.x;
    const int e    = gt >> 5;
    const int lane = gt & 31;
    if (e >= E) return;

    const int s = src[e];
    const int d = dst[e];
    const float nrm = dinv[s] * dinv[d];

    const float4* __restrict__ hv = (const float4*)(H + (size_t)s * HIDDEN);
    const float4 v = hv[lane];

    float* __restrict__ o = out + (size_t)d * HIDDEN + lane * 4;
    atomicAdd(o + 0, v.x * nrm);
    atomicAdd(o + 1, v.y * nrm);
    atomicAdd(o + 2, v.z * nrm);
    atomicAdd(o + 3, v.w * nrm);
}

// ---------------- launcher ----------------

extern "C" void kernel_launch(void* const* d_in, const int* in_sizes, int n_in,
                              void* d_out, int out_size, void* d_ws, size_t ws_size,
                              hipStream_t stream) {
    (void)in_sizes; (void)n_in; (void)out_size; (void)ws_size;

    const float* x    = (const float*)d_in[0];                 // [N, 128]
    const int*   edge = (const int*)  d_in[1];                 // [2, E]
    const float* Ws   = (const float*)d_in[2];                 // [2, 128, 128]
    const float* bs   = (const float*)d_in[3];                 // [2, 128]
    float*       out  = (float*)d_out;                         // [N, 128]

    const int* srcIdx = edge;             // edge_index[0]
    const int* dstIdx = edge + N_EDGES;   // edge_index[1]

    // workspace layout (all fp32): dinv[N] | h[N*128] | x1[N*128]  (~103 MB)
    float* ws   = (float*)d_ws;
    float* dinv = ws;
    float* h    = ws + N_NODES;
    float* x1   = h + (size_t)N_NODES * HIDDEN;

    const dim3 blk(256);
    const int totalNF = N_NODES * HIDDEN;                       // 12.8M
    const long long scatterThreads = (long long)N_EDGES * 32;   // 51.2M

    // edge normalization: deg -> dinv
    zero_f32      <<<(N_NODES + 255) / 256, blk, 0, stream>>>(dinv, N_NODES);
    degree_kernel <<<(N_EDGES + 255) / 256, blk, 0, stream>>>(dstIdx, dinv, N_EDGES);
    dinv_kernel   <<<(N_NODES + 255) / 256, blk, 0, stream>>>(dinv, N_NODES);

    // ---- layer 0: h = x @ Ws[0]; x1 = scatter(h) + bs[0] ----
    gemm_wmma_f32   <<<N_NODES / 16, blk, 0, stream>>>(x, Ws, h);
    bias_init_kernel<<<(totalNF + 255) / 256, blk, 0, stream>>>(bs, x1, totalNF);
    scatter_kernel  <<<(int)((scatterThreads + 255) / 256), blk, 0, stream>>>(
        h, srcIdx, dstIdx, dinv, x1, N_EDGES);

    // ---- layer 1: h = x1 @ Ws[1]; out = scatter(h) + bs[1] ----
    gemm_wmma_f32   <<<N_NODES / 16, blk, 0, stream>>>(x1, Ws + HIDDEN * HIDDEN, h);
    bias_init_kernel<<<(totalNF + 255) / 256, blk, 0, stream>>>(bs + HIDDEN, out, totalNF);
    scatter_kernel  <<<(int)((scatterThreads + 255) / 256), blk, 0, stream>>>(
        h, srcIdx, dstIdx, dinv, out, N_EDGES);
}